// CTCLayer_3358664425530
// MI455X (gfx1250) — compile-verified
//
#include <hip/hip_runtime.h>
#include <hip/hip_bf16.h>

#define B_ 512
#define T_ 512
#define C_ 256
#define L_ 64
#define S_ (2 * L_ + 1)       /* 129 extended states */
#define BLK 160               /* 5 wave32 waves */
#define RING 8                /* row ring buffers (power of two) */
#define DEPTH 6               /* async prefetch distance, <= RING-2 */
#define NEGF (-1e30f)
#define EPSF (1e-7f)

// ---------------------------------------------------------------------------
// CDNA5 async global->LDS copy (ASYNCcnt-tracked DMA) + counter wait.
// Builtin signature (confirmed by hipcc): (v4i addrspace(1)*, v4i
// addrspace(3)*, imm offset, imm cpol).
// ---------------------------------------------------------------------------
typedef int v4i __attribute__((vector_size(4 * sizeof(int))));
typedef __attribute__((address_space(1))) v4i* gv4i_p;   // global v4i*
typedef __attribute__((address_space(3))) v4i* lv4i_p;   // LDS    v4i*

__device__ __forceinline__ void async_copy_b128(const float* g, float* l) {
#if __has_builtin(__builtin_amdgcn_global_load_async_to_lds_b128)
  __builtin_amdgcn_global_load_async_to_lds_b128(
      (gv4i_p)(g), (lv4i_p)(l), /*offset=*/0, /*cpol=*/0);
#else
  unsigned lds_addr =
      (unsigned)(__UINTPTR_TYPE__)(__attribute__((address_space(3))) void*)l;
  asm volatile("global_load_async_to_lds_b128 %0, %1, off"
               :: "v"(lds_addr), "v"(g) : "memory");
#endif
}

#if __has_builtin(__builtin_amdgcn_s_wait_asynccnt)
#define WAIT_ASYNC(n) __builtin_amdgcn_s_wait_asynccnt(n)
#else
#define WAIT_ASYNC(n) asm volatile("s_wait_asynccnt " #n ::: "memory")
#endif

// Stable 3-way logaddexp; NEGF acts as -inf (exp underflows to 0).
__device__ __forceinline__ float logaddexp3(float a, float b, float c) {
  float m = fmaxf(a, fmaxf(b, c));
  return m + __logf(__expf(a - m) + __expf(b - m) + __expf(c - m));
}

__global__ __launch_bounds__(BLK) void ctc_loss_kernel(
    const int* __restrict__ y_true,    // [B, L] int32, labels in [0, C-2]
    const float* __restrict__ y_pred,  // [B, T, C] float32 probabilities
    float* __restrict__ out)           // [B] float32 loss
{
  __shared__ float row[RING][C_];      // ring of probability rows (8 KB)
  __shared__ float alpha[2][S_ + 2];   // ping-pong alpha, 2 NEG guard slots
  __shared__ int   labels[L_];

  const int  tid    = threadIdx.x;
  const int  b      = blockIdx.x;
  const bool loader = (tid < 64);      // waves 0-1: 64 lanes x 16 B = 1 KB/row

  if (tid < L_) labels[tid] = y_true[b * L_ + tid];
  if (tid < 2)  { alpha[0][tid] = NEGF; alpha[1][tid] = NEGF; }
  __syncthreads();

  // Per-state constants: class index + skip-transition flag.
  const int s = tid;
  int  cls  = C_ - 1;                  // blank for even states
  bool skip = false;
  if (s < S_ && (s & 1)) {
    const int j = s >> 1;
    cls = labels[j];
    if (s >= 3) skip = (labels[j] != labels[j - 1]);
  }

  const float* rowbase = y_pred + (size_t)b * T_ * C_;

  // Preload rows 0..DEPTH-1.
  if (loader) {
#pragma unroll
    for (int r = 0; r < DEPTH; ++r)
      async_copy_b128(rowbase + (size_t)r * C_ + tid * 4, &row[r][tid * 4]);
  }

  // ---- t = 0 (peeled): issue row DEPTH, retire row 0, init alpha ----------
  if (loader)
    async_copy_b128(rowbase + (size_t)DEPTH * C_ + tid * 4,
                    &row[DEPTH & (RING - 1)][tid * 4]);
  WAIT_ASYNC(DEPTH);      // DEPTH+1 issued; <=DEPTH outstanding => row 0 done
  __syncthreads();        // publish row 0 to all waves

  if (s < S_) {
    float lp = __logf(row[0][cls] + EPSF);
    alpha[0][2 + s] = (s < 2) ? lp : NEGF;  // only states 0,1 reachable at t=0
  }
  int cur = 0;

  // ---- main scan -----------------------------------------------------------
  for (int t = 1; t < T_; ++t) {
    // Prefetch row t+DEPTH into ring slot (t+DEPTH)&7. That slot was last
    // read at step t-2, ordered before this issue by the step t-1 barrier.
    if (t + DEPTH < T_ && loader)
      async_copy_b128(rowbase + (size_t)(t + DEPTH) * C_ + tid * 4,
                      &row[(t + DEPTH) & (RING - 1)][tid * 4]);
    if (t + DEPTH < T_) {
      WAIT_ASYNC(DEPTH);  // steady state: rows t+1..t+DEPTH in flight
    } else {
      WAIT_ASYNC(0);      // drain tail: ensures row t landed
    }
    __syncthreads();      // row t + alpha[cur] visible to all waves

    if (s < S_) {
      float lp = __logf(row[t & (RING - 1)][cls] + EPSF);
      float a  = alpha[cur][2 + s];
      float a1 = alpha[cur][1 + s];            // guard slot gives NEG at s=0
      float a2 = skip ? alpha[cur][s] : NEGF;  // guard slots cover s<2
      alpha[cur ^ 1][2 + s] = logaddexp3(a, a1, a2) + lp;
    }
    cur ^= 1;
  }

  __syncthreads();
  if (tid == 0) {
    float x = alpha[cur][2 + S_ - 1];
    float y = alpha[cur][2 + S_ - 2];
    float m = fmaxf(x, y);
    out[b] = -(m + __logf(__expf(x - m) + __expf(y - m)));
  }
}

extern "C" void kernel_launch(void* const* d_in, const int* in_sizes, int n_in,
                              void* d_out, int out_size, void* d_ws, size_t ws_size,
                              hipStream_t stream) {
  (void)in_sizes; (void)n_in; (void)out_size; (void)d_ws; (void)ws_size;
  const int*   y_true = (const int*)d_in[0];    // [B, L]
  const float* y_pred = (const float*)d_in[1];  // [B, T, C]
  float*       out    = (float*)d_out;          // [B, 1]
  ctc_loss_kernel<<<dim3(B_), dim3(BLK), 0, stream>>>(y_true, y_pred, out);
}